// GConv_27178553049849
// MI455X (gfx1250) — compile-verified
//
#include <hip/hip_runtime.h>
#include <hip/hip_bf16.h>

typedef float v2f __attribute__((ext_vector_type(2)));
typedef float v8f __attribute__((ext_vector_type(8)));

__device__ __forceinline__ v8f wmma4(v2f a, v2f b, v8f c) {
  // V_WMMA_F32_16X16X4_F32 : D = A(16x4) x B(4x16) + C(16x16), fp32
  return __builtin_amdgcn_wmma_f32_16x16x4_f32(false, a, false, b, (short)0, c, false, false);
}

// ---------------------------------------------------------------- zero fill
__global__ void k_zero(float4* __restrict__ p, long long n4) {
  long long i = (long long)blockIdx.x * blockDim.x + threadIdx.x;
  if (i < n4) p[i] = make_float4(0.f, 0.f, 0.f, 0.f);
}

// ---------------------------------------------------------------- EdgeConv (fused MLP + atomic-max aggregate)
#define EC_WAVES 4
__global__ __launch_bounds__(128) void k_edgeconv(
    const float* __restrict__ x, const int* __restrict__ src, const int* __restrict__ dst,
    const float* __restrict__ W1, const float* __restrict__ b1, const float* __restrict__ g1, const float* __restrict__ t1,
    const float* __restrict__ W2, const float* __restrict__ b2, const float* __restrict__ g2, const float* __restrict__ t2,
    const float* __restrict__ W3, const float* __restrict__ b3, const float* __restrict__ g3, const float* __restrict__ t3,
    float* __restrict__ out /* concat base, 1024 cols, writes cols 0..63 */, int E, int ntiles)
{
  __shared__ __align__(16) float Wf1[64 * 8];
  __shared__ __align__(16) float Wf2[64 * 64];
  __shared__ __align__(16) float Wf3[64 * 64];
  __shared__ float bf1[64], bf2[64], bf3[64];
  __shared__ __align__(16) float act[EC_WAVES][16 * 64];
  __shared__ int dsts[EC_WAVES][16];

  const int tid = threadIdx.x;
  for (int i = tid; i < 64 * 8; i += blockDim.x) { int n = i >> 3; Wf1[i] = W1[i] * g1[n]; }
  for (int i = tid; i < 64 * 64; i += blockDim.x) {
    int n = i >> 6;
    Wf2[i] = W2[i] * g2[n];
    Wf3[i] = W3[i] * g3[n];
  }
  for (int i = tid; i < 64; i += blockDim.x) {
    bf1[i] = b1[i] * g1[i] + t1[i];
    bf2[i] = b2[i] * g2[i] + t2[i];
    bf3[i] = b3[i] * g3[i] + t3[i];
  }
  __syncthreads();

  const int wave = tid >> 5, lane = tid & 31;
  const int mrow = lane & 15, half = lane >> 4;
  const int tile = blockIdx.x * EC_WAVES + wave;
  if (tile >= ntiles) return;               // wave-uniform
  const int e0 = tile * 16;
  float* A = act[wave];

  // ---- gather: m = [x_dst | x_src - x_dst] into A[16][cols 0..7] (row stride 64)
  {
    const int e = e0 + mrow;
    const int d = dst[e];
    const float4 xi = *(const float4*)(x + (size_t)d * 4);
    if (half == 0) {
      dsts[wave][mrow] = d;
      *(float4*)&A[mrow * 64 + 0] = xi;
    } else {
      const int s = src[e];
      const float4 xj = *(const float4*)(x + (size_t)s * 4);
      *(float4*)&A[mrow * 64 + 4] = make_float4(xj.x - xi.x, xj.y - xi.y, xj.z - xi.z, xj.w - xi.w);
    }
  }
  // same-wave LDS ops are in-order (hw guarantee); loads below see the stores.

  // ---- layer 1: [16x8] @ [8x64]
  v2f a10 = *(const v2f*)&A[mrow * 64 + 0 + 2 * half];
  v2f a11 = *(const v2f*)&A[mrow * 64 + 4 + 2 * half];
#pragma unroll
  for (int nt = 0; nt < 4; ++nt) {
    const int n = nt * 16 + mrow;
    const float bb = bf1[n];
    v8f c = {bb, bb, bb, bb, bb, bb, bb, bb};
    v2f b = *(const v2f*)&Wf1[n * 8 + 0 + 2 * half];
    c = wmma4(a10, b, c);
    b = *(const v2f*)&Wf1[n * 8 + 4 + 2 * half];
    c = wmma4(a11, b, c);
#pragma unroll
    for (int r = 0; r < 8; ++r) {
      float v = c[r]; v = v > 0.f ? v : 0.f;
      A[(r + 8 * half) * 64 + n] = v;
    }
  }

  // ---- layer 2: [16x64] @ [64x64]
  {
    v2f a[16];
#pragma unroll
    for (int j = 0; j < 16; ++j) a[j] = *(const v2f*)&A[mrow * 64 + 4 * j + 2 * half];
#pragma unroll
    for (int nt = 0; nt < 4; ++nt) {
      const int n = nt * 16 + mrow;
      const float bb = bf2[n];
      v8f c = {bb, bb, bb, bb, bb, bb, bb, bb};
#pragma unroll
      for (int j = 0; j < 16; ++j) {
        v2f b = *(const v2f*)&Wf2[n * 64 + 4 * j + 2 * half];
        c = wmma4(a[j], b, c);
      }
#pragma unroll
      for (int r = 0; r < 8; ++r) {
        float v = c[r]; v = v > 0.f ? v : 0.f;
        A[(r + 8 * half) * 64 + n] = v;
      }
    }
  }

  // ---- layer 3: [16x64] @ [64x64], relu, atomic-max scatter into out[dst][0..63]
  {
    v2f a[16];
#pragma unroll
    for (int j = 0; j < 16; ++j) a[j] = *(const v2f*)&A[mrow * 64 + 4 * j + 2 * half];
#pragma unroll
    for (int nt = 0; nt < 4; ++nt) {
      const int n = nt * 16 + mrow;
      const float bb = bf3[n];
      v8f c = {bb, bb, bb, bb, bb, bb, bb, bb};
#pragma unroll
      for (int j = 0; j < 16; ++j) {
        v2f b = *(const v2f*)&Wf3[n * 64 + 4 * j + 2 * half];
        c = wmma4(a[j], b, c);
      }
#pragma unroll
      for (int r = 0; r < 8; ++r) {
        float v = c[r]; v = v > 0.f ? v : 0.f;
        const int d = dsts[wave][r + 8 * half];
        atomicMax((unsigned int*)&out[(size_t)d * 1024 + n], __float_as_uint(v));
      }
    }
  }
}

// ---------------------------------------------------------------- per-edge scatter-add (segment_sum of h[src] at dst)
__global__ void k_edge_agg(const int* __restrict__ src, const int* __restrict__ dst,
                           const float* __restrict__ hin, int ldh,
                           float* __restrict__ agg, int C, int c4shift, long long total)
{
  long long idx = (long long)blockIdx.x * blockDim.x + threadIdx.x;
  if (idx >= total) return;
  const int e = (int)(idx >> c4shift);
  const int c = (int)((idx & ((1 << c4shift) - 1)) << 2);
  const int s = src[e], d = dst[e];
  const float4 v = *(const float4*)(hin + (size_t)s * ldh + c);
  float* p = agg + (size_t)d * C + c;
  atomicAdd(p + 0, v.x);
  atomicAdd(p + 1, v.y);
  atomicAdd(p + 2, v.z);
  atomicAdd(p + 3, v.w);
}

// ---------------------------------------------------------------- generic WMMA GEMM:
//   out = epilogue( A0[M,K0] @ W0[N,K0]^T + A1[M,K1] @ W1[N,K1]^T + bias )
//   mode 0: none   mode 1: relu then *g+t   mode 2: *g+t then relu
__global__ __launch_bounds__(128) void k_gemm(
    const float* __restrict__ A0, int lda0, int K0,
    const float* __restrict__ A1, int lda1, int K1,
    const float* __restrict__ W0, const float* __restrict__ W1,
    const float* __restrict__ bias, const float* __restrict__ gma, const float* __restrict__ bta,
    float* __restrict__ out, int ldo, int M, int Nc, int mode)
{
  const int tid = threadIdx.x, wave = tid >> 5, lane = tid & 31;
  const int mrow = lane & 15, half = lane >> 4;
  const int n0 = (blockIdx.x * 4 + wave) * 16;
  if (n0 >= ((Nc + 15) & ~15)) return;      // wave-uniform
  const int m0 = blockIdx.y * 16;
  const int n = n0 + mrow;
  const int ncl = n < Nc ? n : Nc - 1;
  const int m = m0 + mrow;                  // M is a multiple of 16

  const float bb = bias ? bias[ncl] : 0.f;
  v8f c = {bb, bb, bb, bb, bb, bb, bb, bb};

  for (int kb = 0; kb < K0; kb += 4) {
    v2f a = *(const v2f*)(A0 + (size_t)m * lda0 + kb + 2 * half);
    v2f b = *(const v2f*)(W0 + (size_t)ncl * K0 + kb + 2 * half);
    c = wmma4(a, b, c);
  }
  for (int kb = 0; kb < K1; kb += 4) {
    v2f a = *(const v2f*)(A1 + (size_t)m * lda1 + kb + 2 * half);
    v2f b = *(const v2f*)(W1 + (size_t)ncl * K1 + kb + 2 * half);
    c = wmma4(a, b, c);
  }

  float g = 1.f, t = 0.f;
  if (gma) { g = gma[ncl]; t = bta[ncl]; }
#pragma unroll
  for (int r = 0; r < 8; ++r) {
    float v = c[r];
    if (mode == 1) { v = fmaxf(v, 0.f); v = v * g + t; }
    else if (mode == 2) { v = v * g + t; v = fmaxf(v, 0.f); }
    const int row = m0 + r + 8 * half;
    if (n < Nc) out[(size_t)row * ldo + n] = v;
  }
}

// ---------------------------------------------------------------- per-graph max + mean pool (contiguous segments)
__global__ void k_pool(const float* __restrict__ cc, float* __restrict__ z, int S) {
  const int c = blockIdx.x * blockDim.x + threadIdx.x;   // 0..1023
  const int g = blockIdx.y;
  const float* p = cc + (size_t)g * S * 1024 + c;
  float mx = -3.402823466e38f, sm = 0.f;
  for (int i = 0; i < S; ++i) {
    const float v = p[(size_t)i * 1024];
    mx = fmaxf(mx, v);
    sm += v;
  }
  z[(size_t)g * 2048 + c] = mx;
  z[(size_t)g * 2048 + 1024 + c] = sm / (float)S;
}

// ---------------------------------------------------------------- log_softmax over last dim
__global__ void k_logsoftmax(const float* __restrict__ logits, float* __restrict__ out, int G, int C) {
  const int r = blockIdx.x * blockDim.x + threadIdx.x;
  if (r >= G) return;
  const float* p = logits + (size_t)r * C;
  float m = -3.402823466e38f;
  for (int j = 0; j < C; ++j) m = fmaxf(m, p[j]);
  float s = 0.f;
  for (int j = 0; j < C; ++j) s += expf(p[j] - m);
  const float ls = logf(s);
  for (int j = 0; j < C; ++j) out[(size_t)r * C + j] = p[j] - m - ls;
}

// ================================================================ launch
extern "C" void kernel_launch(void* const* d_in, const int* in_sizes, int n_in,
                              void* d_out, int out_size, void* d_ws, size_t ws_size,
                              hipStream_t stream) {
  const float* x = (const float*)d_in[0];
  const int* ei = (const int*)d_in[1];
  // d_in[2] = batch (unused: segments are contiguous, S nodes per graph)

  int p = 3;
  const float *ecW[3], *ecB[3], *ecG[3], *ecT[3];
  for (int i = 0; i < 3; ++i) {
    ecW[i] = (const float*)d_in[p++]; ecB[i] = (const float*)d_in[p++];
    ecG[i] = (const float*)d_in[p++]; ecT[i] = (const float*)d_in[p++];
  }
  const float *gWr[4], *gWn[4], *gB[4], *gG[4], *gT[4];
  for (int i = 0; i < 4; ++i) {
    gWr[i] = (const float*)d_in[p++]; gWn[i] = (const float*)d_in[p++];
    gB[i] = (const float*)d_in[p++];  gG[i] = (const float*)d_in[p++];
    gT[i] = (const float*)d_in[p++];
  }
  const float* W1h = (const float*)d_in[p++]; const float* b1h = (const float*)d_in[p++];
  const float* g1h = (const float*)d_in[p++]; const float* t1h = (const float*)d_in[p++];
  const float* W2h = (const float*)d_in[p++]; const float* b2h = (const float*)d_in[p++];
  const float* g2h = (const float*)d_in[p++]; const float* t2h = (const float*)d_in[p++];
  const float* W3h = (const float*)d_in[p++]; const float* b3h = (const float*)d_in[p++];

  const size_t N = (size_t)in_sizes[0] / 4;
  const size_t E = (size_t)in_sizes[1] / 2;
  const int G = out_size / 40;
  const int S = (int)(N / (size_t)G);

  float* ws = (float*)d_ws;
  float* concat = ws;                               // [N,1024]
  float* agg = concat + N * 1024;                   // [N,<=256]
  float* z = agg + N * 256;                         // [G,2048]
  float* h1 = z + (size_t)G * 2048;                 // [G,512]
  float* h2 = h1 + (size_t)G * 512;                 // [G,128]
  float* logits = h2 + (size_t)G * 128;             // [G,40]

  // zero concat (atomic-max target for EdgeConv; isolated nodes -> 0)
  {
    long long n4 = (long long)(N * 1024 / 4);
    k_zero<<<(unsigned)((n4 + 255) / 256), 256, 0, stream>>>((float4*)concat, n4);
  }

  // EdgeConv -> concat cols [0,64)
  {
    const int ntiles = (int)(E / 16);
    k_edgeconv<<<(ntiles + EC_WAVES - 1) / EC_WAVES, 128, 0, stream>>>(
        x, ei, ei + E,
        ecW[0], ecB[0], ecG[0], ecT[0],
        ecW[1], ecB[1], ecG[1], ecT[1],
        ecW[2], ecB[2], ecG[2], ecT[2],
        concat, (int)E, ntiles);
  }

  // GraphConv layers (reassociated: aggregate features first, then one GEMM)
  const int ci_[4] = {64, 64, 128, 256};
  const int co_[4] = {64, 128, 256, 512};
  const int cin0_[4] = {0, 64, 128, 256};
  const int cout0_[4] = {64, 128, 256, 512};
  for (int l = 0; l < 4; ++l) {
    const int ci = ci_[l], co = co_[l];
    long long an4 = (long long)(N * ci / 4);
    k_zero<<<(unsigned)((an4 + 255) / 256), 256, 0, stream>>>((float4*)agg, an4);

    const long long tot = (long long)E * (ci / 4);
    const int shift = __builtin_ctz(ci / 4);
    k_edge_agg<<<(unsigned)((tot + 255) / 256), 256, 0, stream>>>(
        ei, ei + E, concat + cin0_[l], 1024, agg, ci, shift, tot);

    dim3 grid((co / 16 + 3) / 4, (unsigned)(N / 16));
    k_gemm<<<grid, 128, 0, stream>>>(
        concat + cin0_[l], 1024, ci,
        agg, ci, ci,
        gWr[l], gWn[l], gB[l], gG[l], gT[l],
        concat + cout0_[l], 1024, (int)N, co, /*mode=*/1);
  }

  // global max + mean pool -> z [G,2048]
  k_pool<<<dim3(1024 / 256, G), 256, 0, stream>>>(concat, z, S);

  // head
  k_gemm<<<dim3((512 / 16 + 3) / 4, G / 16), 128, 0, stream>>>(
      z, 2048, 2048, nullptr, 0, 0, W1h, nullptr, b1h, g1h, t1h, h1, 512, G, 512, /*mode=*/2);
  k_gemm<<<dim3((128 / 16 + 3) / 4, G / 16), 128, 0, stream>>>(
      h1, 512, 512, nullptr, 0, 0, W2h, nullptr, b2h, g2h, t2h, h2, 128, G, 128, /*mode=*/2);
  k_gemm<<<dim3(1, G / 16), 128, 0, stream>>>(
      h2, 128, 128, nullptr, 0, 0, W3h, nullptr, b3h, nullptr, nullptr, logits, 40, G, 40, /*mode=*/0);

  k_logsoftmax<<<(G + 127) / 128, 128, 0, stream>>>(logits, (float*)d_out, G, 40);
}